// RoIPooling_33449205301965
// MI455X (gfx1250) — compile-verified
//
#include <hip/hip_runtime.h>
#include <hip/hip_bf16.h>

#define RP_P 7
#define RP_H 50
#define RP_W 50
#define RP_CPB 4   // channels per block

typedef __attribute__((ext_vector_type(4))) unsigned int u32x4;
typedef __attribute__((ext_vector_type(8))) int          i32x8;
typedef __attribute__((ext_vector_type(4))) int          i32x4;

__global__ __launch_bounds__(256) void roi_pool_tdm_kernel(
    const float* __restrict__ x,      // (1,256,50,50)
    const float* __restrict__ rois,   // (N,4)
    float* __restrict__ out,          // (N*C,7,7)
    int C) {
  const int n   = blockIdx.x;
  const int c0  = blockIdx.y * RP_CPB;
  const int tid = threadIdx.x;

  __shared__ float tile[RP_CPB * RP_H * RP_W];   // 40 KB, worst-case tile

  // Integer RoI box: replicates (rois * 0.0625f).astype(int32); rois >= 0 so
  // C truncation == jnp truncation. 50/800 == 0.0625 exactly in f32.
  const float sc = 0.0625f;
  const int x0 = (int)(rois[4 * n + 0] * sc);
  const int y0 = (int)(rois[4 * n + 1] * sc);
  const int x1 = (int)(rois[4 * n + 2] * sc);
  const int y1 = (int)(rois[4 * n + 3] * sc);
  const int rh = y1 - y0 + 1;
  const int rw = x1 - x0 + 1;

  // Clamped staging footprint; every (clamped) row/col the bins touch lies in it.
  const int rowlo = min(max(y0, 0), RP_H - 1);
  const int rowhi = min(max(y1, 0), RP_H - 1);
  const int collo = min(max(x0, 0), RP_W - 1);
  const int colhi = min(max(x1, 0), RP_W - 1);
  const int tileH = rowhi - rowlo + 1;
  const int tileW = colhi - collo + 1;

  // --- Stage (tileW x tileH x CPB) sub-tensor into LDS via the Tensor Data Mover.
  // One TDM op per block; EXEC is ignored by TENSOR_LOAD_TO_LDS so issuing it
  // from lane 0 of wave 0 moves the whole tile. Other waves branch around it.
  if (tid == 0) {
    const unsigned lds_off = (unsigned)(uintptr_t)(void*)tile;  // LDS aperture: low 32 bits = byte offset
    const unsigned long long gaddr =
        (unsigned long long)(uintptr_t)(x + ((size_t)c0 * (RP_H * RP_W) +
                                             (size_t)rowlo * RP_W + (size_t)collo));
    u32x4 g0;
    i32x8 g1;
    i32x4 g2, g3;
    i32x8 gx = {0, 0, 0, 0, 0, 0, 0, 0};   // extra operand (clang-23 form); reserved -> 0

    // D# group 0: count=1 (valid), gather off | lds_addr | global_addr | type=2
    g0[0] = 1u;
    g0[1] = lds_off;
    g0[2] = (unsigned)(gaddr & 0xFFFFFFFFull);
    g0[3] = (unsigned)((gaddr >> 32) & 0x01FFFFFFull) | (2u << 30);

    // D# group 1: data_size=2 (4B); tensor_dim0=tileW, tensor_dim1=tileH;
    // tile_dims = (tileW, tileH, CPB); strides: dim0=50 elems, dim1=2500 elems.
    g1[0] = (int)(2u << 16);                                         // data_size=4B, no pad/iterate/mask
    g1[1] = (int)(((unsigned)tileW & 0xFFFFu) << 16);                // tensor_dim0[15:0]
    g1[2] = (int)((((unsigned)tileW >> 16) & 0xFFFFu) |
                  (((unsigned)tileH & 0xFFFFu) << 16));              // dim0 hi / dim1 lo
    g1[3] = (int)((((unsigned)tileH >> 16) & 0xFFFFu) |
                  (((unsigned)tileW & 0xFFFFu) << 16));              // dim1 hi / tile_dim0
    g1[4] = (int)(((unsigned)tileH & 0xFFFFu) |
                  ((unsigned)RP_CPB << 16));                         // tile_dim1 / tile_dim2
    g1[5] = RP_W;                                                    // tensor_dim0_stride lo32 (=50)
    g1[6] = (int)(((unsigned)(RP_H * RP_W) & 0xFFFFu) << 16);        // stride0 hi=0 / stride1 lo16 (=2500)
    g1[7] = (int)((unsigned)(RP_H * RP_W) >> 16);                    // stride1 bits[47:16] (=0)

    // D# group 2: tensor_dim2 = CPB (3D tile); dims 3/4 unused (tile_dim3=0).
    g2[0] = RP_CPB;
    g2[1] = 1;
    g2[2] = 0;
    g2[3] = 0;
    // D# group 3: unused higher dims.
    g3[0] = 0; g3[1] = 0; g3[2] = 0; g3[3] = 0;

    __builtin_amdgcn_tensor_load_to_lds(g0, g1, g2, g3, gx, 0);
  }
  __builtin_amdgcn_s_wait_tensorcnt(0);   // no-op for waves that issued nothing
  __syncthreads();

  // --- Max-pool out of LDS: 4 channels * 49 bins = 196 active threads.
  if (tid < RP_CPB * (RP_P * RP_P)) {
    const int c   = tid / (RP_P * RP_P);
    const int bin = tid % (RP_P * RP_P);
    const int ph  = bin / RP_P;
    const int pw  = bin % RP_P;

    const int hs = y0 + (ph * rh) / RP_P;
    const int he = y0 + ((ph + 1) * rh + RP_P - 1) / RP_P;
    const int wl = x0 + (pw * rw) / RP_P;
    const int we = x0 + ((pw + 1) * rw + RP_P - 1) / RP_P;

    float v = -__builtin_inff();
    const float* __restrict__ cbase = tile + c * (tileH * tileW);
    for (int r = hs; r < he; ++r) {
      int rc = min(max(r, 0), RP_H - 1) - rowlo;
      rc = min(max(rc, 0), tileH - 1);                 // defensive; analytically in range
      const float* __restrict__ rp = cbase + rc * tileW;
      for (int col = wl; col < we; ++col) {
        int cc = min(max(col, 0), RP_W - 1) - collo;
        cc = min(max(cc, 0), tileW - 1);
        v = fmaxf(v, rp[cc]);
      }
    }
    out[(((size_t)n * C + (size_t)(c0 + c)) * RP_P + ph) * RP_P + pw] = v;
  }
}

extern "C" void kernel_launch(void* const* d_in, const int* in_sizes, int n_in,
                              void* d_out, int out_size, void* d_ws, size_t ws_size,
                              hipStream_t stream) {
  const float* x    = (const float*)d_in[0];   // (1,256,50,50) f32
  // d_in[1] = img, only its (fixed) shape matters -> scale hardcoded 0.0625f
  const float* rois = (const float*)d_in[2];   // (N,4) f32
  float* out = (float*)d_out;

  const int N = in_sizes[2] / 4;                 // 128
  const int C = in_sizes[0] / (RP_H * RP_W);     // 256

  dim3 grid((unsigned)N, (unsigned)(C / RP_CPB));  // 128 x 64 blocks
  roi_pool_tdm_kernel<<<grid, 256, 0, stream>>>(x, rois, out, C);
}